// MaxPool_35304631174310
// MI455X (gfx1250) — compile-verified
//
#include <hip/hip_runtime.h>
#include <stdint.h>

// ---------------------------------------------------------------------------
// GraphSAGE-pool layer for MI455X (gfx1250, wave32, WMMA).
//   h      = x @ W_node + b_node                       -> out[:,   0:256]
//   h_agg  = x @ W_agg  + b_agg                        -> ws
//   agg    = segment_max(h_agg[src], dst) (zero-fill)  -> ws (ordered u32)
//   neighs = agg @ W_neigh + b_neigh                   -> out[:, 256:512]
//   out    = l2_normalize(out, axis=-1)
// All GEMMs use V_WMMA_F32_16X16X4_F32 (exact fp32, matches reference dtype).
// K/N are compile-time so all inner-loop addressing folds into load-immediate
// offsets (IOFFSET) instead of per-iteration 64-bit VALU address math.
// ---------------------------------------------------------------------------

typedef __attribute__((ext_vector_type(2))) float v2f;
typedef __attribute__((ext_vector_type(8))) float v8f;

#define DFEAT  256
#define HIDDEN 16
#define OUTD   256
#define LDC    512   // concat width

// Order-preserving f32 <-> u32 map so fp-max == unsigned-int max.
// Any finite/-inf float maps to u > 0, so u==0 is a safe "no in-edge" sentinel.
__device__ __forceinline__ unsigned f32_to_ord(float f) {
  unsigned b = __float_as_uint(f);
  return (b & 0x80000000u) ? ~b : (b | 0x80000000u);
}
__device__ __forceinline__ float ord_to_f32(unsigned u) {
  return (u & 0x80000000u) ? __uint_as_float(u ^ 0x80000000u)
                           : __uint_as_float(~u);
}

// ---------------------------------------------------------------------------
// One wave32 computes a 16x16 tile of C = A[MxK] * W[KxN] + bias using
// V_WMMA_F32_16X16X4_F32, stepping K by 4.
// A-frag (16x4 f32, 2 VGPR): lane L holds row M=L&15, K = (L>>4)*2 + {0,1}.
// B-frag (4x16 f32, 2 VGPR): lane L holds col N=L&15, K = (L>>4)*2 + {0,1}.
// C/D (16x16 f32, 8 VGPR):   VGPR i = row (i + 8*(L>>4)), col L&15.
// DECODE=true: A is the ordered-u32 max buffer; 0 sentinel -> 0.0f.
// ---------------------------------------------------------------------------
template <bool DECODE, int K, int N>
__global__ __launch_bounds__(256)
void wmma_gemm_bias(const void* __restrict__ Araw,
                    const float* __restrict__ W,
                    const float* __restrict__ bias,
                    float* __restrict__ C,
                    int ldc, int col_off, int tiles_x, int total_tiles) {
  const int wave = (int)((blockIdx.x * blockDim.x + threadIdx.x) >> 5);
  if (wave >= total_tiles) return;
  const int lane = threadIdx.x & 31;
  const int trow = wave / tiles_x;
  const int tcol = wave - trow * tiles_x;
  const int m0 = trow << 4;
  const int n0 = tcol << 4;

  const int r    = lane & 15;          // A-row / B-col / C-col within tile
  const int kb   = (lane >> 4) << 1;   // K sub-offset: 0 or 2
  const int arow = m0 + r;
  const int col  = n0 + r;

  v8f acc;
  const float bv = bias[col];
#pragma unroll
  for (int i = 0; i < 8; ++i) acc[i] = bv;

  // Per-lane base pointers; all per-k deltas are compile-time immediates.
  const float*    Ap = (const float*)Araw + (size_t)arow * K + kb;
  const unsigned* Au = (const unsigned*)Araw + (size_t)arow * K + kb;
  const float*    Wp = W + (size_t)kb * N + col;

#pragma unroll 16
  for (int k = 0; k < K; k += 4) {
    v2f a, b;
    if constexpr (DECODE) {
      const unsigned u0 = Au[k];
      const unsigned u1 = Au[k + 1];
      a.x = (u0 == 0u) ? 0.0f : ord_to_f32(u0);
      a.y = (u1 == 0u) ? 0.0f : ord_to_f32(u1);
    } else {
      a = *(const v2f*)(Ap + k);               // global_load_b64, imm offset
    }
    b.x = Wp[(size_t)k * N];                   // global_load_b32, imm offset
    b.y = Wp[(size_t)(k + 1) * N];
    acc = __builtin_amdgcn_wmma_f32_16x16x4_f32(
        /*neg_a=*/false, a, /*neg_b=*/false, b,
        /*c_mod=*/(short)0, acc, /*reuse_a=*/false, /*reuse_b=*/false);
  }

  float* cp = C + (size_t)(m0 + ((lane >> 4) << 3)) * ldc + col_off + col;
#pragma unroll
  for (int i = 0; i < 8; ++i) cp[(size_t)i * ldc] = acc[i];
}

// Zero-init the ordered-u32 max buffer (sentinel 0 == "no in-edge").
__global__ void init_u32_kernel(unsigned* __restrict__ p, int n) {
  int i = blockIdx.x * blockDim.x + threadIdx.x;
  if (i < n) p[i] = 0u;
}

// One thread per edge: gather 16 floats of h_agg[src], atomicMax into agg[dst].
// agg table is 6.4 MB -> L2-resident; u32 max is exact & order-independent.
__global__ __launch_bounds__(256)
void scatter_max_kernel(const int* __restrict__ src, const int* __restrict__ dst,
                        const float* __restrict__ h_agg,
                        unsigned* __restrict__ agg, int E) {
  int e = blockIdx.x * blockDim.x + threadIdx.x;
  if (e >= E) return;
  const int s = src[e];
  const int d = dst[e];
  const float4* hp = (const float4*)(h_agg + (size_t)s * HIDDEN);
  unsigned* ap = agg + (size_t)d * HIDDEN;
#pragma unroll
  for (int q = 0; q < 4; ++q) {
    float4 v = hp[q];
    atomicMax(ap + q * 4 + 0, f32_to_ord(v.x));
    atomicMax(ap + q * 4 + 1, f32_to_ord(v.y));
    atomicMax(ap + q * 4 + 2, f32_to_ord(v.z));
    atomicMax(ap + q * 4 + 3, f32_to_ord(v.w));
  }
}

// One wave32 per 512-wide row: sum of squares, wave reduce, scale by rsqrt.
__global__ __launch_bounds__(256)
void l2norm_rows_kernel(float* __restrict__ C, int M) {
  const int row = (int)((blockIdx.x * blockDim.x + threadIdx.x) >> 5);
  if (row >= M) return;
  const int lane = threadIdx.x & 31;
  float* rp = C + (size_t)row * LDC;
  float v[16];
  float ss = 0.0f;
#pragma unroll
  for (int j = 0; j < 16; ++j) {
    v[j] = rp[lane + 32 * j];
    ss += v[j] * v[j];
  }
#pragma unroll
  for (int o = 16; o > 0; o >>= 1) ss += __shfl_xor(ss, o, 32);
  const float scale = rsqrtf(fmaxf(ss, 1e-12f));
#pragma unroll
  for (int j = 0; j < 16; ++j) rp[lane + 32 * j] = v[j] * scale;
}

extern "C" void kernel_launch(void* const* d_in, const int* in_sizes, int n_in,
                              void* d_out, int out_size, void* d_ws, size_t ws_size,
                              hipStream_t stream) {
  const float* x       = (const float*)d_in[0];
  const int*   src     = (const int*)d_in[1];
  const int*   dst     = (const int*)d_in[2];
  const float* W_node  = (const float*)d_in[3];
  const float* b_node  = (const float*)d_in[4];
  const float* W_agg   = (const float*)d_in[5];
  const float* b_agg   = (const float*)d_in[6];
  const float* W_neigh = (const float*)d_in[7];
  const float* b_neigh = (const float*)d_in[8];
  float* out = (float*)d_out;

  const int N = in_sizes[0] / DFEAT;   // 100000 (divisible by 16)
  const int E = in_sizes[1];           // 3200000

  // Workspace: h_agg [N x 16] f32, then agg [N x 16] ordered-u32.
  float*    h_agg = (float*)d_ws;
  unsigned* agg   = (unsigned*)((char*)d_ws + (size_t)N * HIDDEN * sizeof(float));

  const int row_tiles = N / 16;

  // 1) h = x @ W_node + b_node  -> out[:, 0:256]
  {
    const int tiles = row_tiles * (OUTD / 16);
    const int blocks = (tiles + 7) / 8;                 // 8 waves / block
    wmma_gemm_bias<false, DFEAT, OUTD><<<blocks, 256, 0, stream>>>(
        x, W_node, b_node, out, LDC, 0, OUTD / 16, tiles);
  }

  // 2) h_agg = x @ W_agg + b_agg -> ws
  {
    const int tiles = row_tiles;                        // HIDDEN/16 == 1
    const int blocks = (tiles + 7) / 8;
    wmma_gemm_bias<false, DFEAT, HIDDEN><<<blocks, 256, 0, stream>>>(
        x, W_agg, b_agg, h_agg, HIDDEN, 0, 1, tiles);
  }

  // 3) agg = 0 sentinel
  {
    const int n = N * HIDDEN;
    init_u32_kernel<<<(n + 255) / 256, 256, 0, stream>>>(agg, n);
  }

  // 4) segment max over edges (ordered-u32 atomicMax)
  scatter_max_kernel<<<(E + 255) / 256, 256, 0, stream>>>(src, dst, h_agg, agg, E);

  // 5) neighs = decode(agg) @ W_neigh + b_neigh -> out[:, 256:512]
  {
    const int tiles = row_tiles * (OUTD / 16);
    const int blocks = (tiles + 7) / 8;
    wmma_gemm_bias<true, HIDDEN, OUTD><<<blocks, 256, 0, stream>>>(
        agg, W_neigh, b_neigh, out, LDC, OUTD, OUTD / 16, tiles);
  }

  // 6) row-wise L2 normalize of out [N x 512]
  {
    const int blocks = (N + 7) / 8;                     // 8 rows / block
    l2norm_rows_kernel<<<blocks, 256, 0, stream>>>(out, N);
  }
}